// SpGraphAttentionLayer_11081015623684
// MI455X (gfx1250) — compile-verified
//
#include <hip/hip_runtime.h>
#include <math.h>

typedef __attribute__((ext_vector_type(2))) float v2f;
typedef __attribute__((ext_vector_type(8))) float v8f;

#define TILE_E 32
#define KDIM   384
#define ROWP   388   // padded LDS row stride (floats): 16B-aligned rows, conflict-free banks
#define CPAD   132   // padded c-tile stride

// ---------------- zero init: out (num accumulator) and denom ----------------
__global__ __launch_bounds__(256) void gat_zero(float* __restrict__ out,
                                                float* __restrict__ denom,
                                                int n_out, int n_den) {
  int i = blockIdx.x * 256 + threadIdx.x;
  int stride = gridDim.x * 256;
  for (int j = i; j < n_out; j += stride) out[j] = 0.f;
  for (int j = i; j < n_den; j += stride) denom[j] = 0.f;
}

// ---------------- fused: gather -> WMMA GEMM -> logits -> exp -> segment atomics ----
__global__ __launch_bounds__(256) void gat_main(
    const float* __restrict__ x,
    const int*   __restrict__ edges,     // [2, E1]
    const float* __restrict__ eembed,    // [E1, 128]
    const int*   __restrict__ nedges,    // [2, E2]
    const float* __restrict__ neembed,   // [E2, 128]
    const float* __restrict__ w1,        // [128, 384]
    const float* __restrict__ w2,        // [1, 128]
    float* __restrict__ out,             // [N, 128]  num accumulator
    float* __restrict__ denom,           // [N]
    int E1, int E2)
{
  __shared__ float       trip[TILE_E * ROWP];   // triple tile, later reused as c tile
  __shared__ int         srcs[TILE_E];
  __shared__ int         dsts[TILE_E];
  __shared__ const float* gbase[TILE_E];
  __shared__ float       partial[TILE_E * 8];
  __shared__ float       bexp[TILE_E];
  __shared__ int         evalid[TILE_E];

  const int tid  = threadIdx.x;
  const int wave = tid >> 5;
  const int lane = tid & 31;
  const int E    = E1 + E2;
  const int base = blockIdx.x * TILE_E;

  // per-edge metadata
  if (tid < TILE_E) {
    int ge = base + tid;
    int s = 0, d = 0, v = 0;
    const float* g = eembed;
    if (ge < E) {
      v = 1;
      if (ge < E1) { s = edges[ge];  d = edges[E1 + ge];  g = eembed  + (size_t)ge * 128; }
      else { int t2 = ge - E1; s = nedges[t2]; d = nedges[E2 + t2]; g = neembed + (size_t)t2 * 128; }
    }
    srcs[tid] = s; dsts[tid] = d; gbase[tid] = g; evalid[tid] = v;
  }
  __syncthreads();

  // cooperative gather of triple tile: 32 edges x 384 floats as float4 (12 per thread)
  #pragma unroll
  for (int i = 0; i < 12; ++i) {
    int idx = tid + i * 256;          // 0..3071
    int el  = idx / 96;
    int q   = idx - el * 96;          // 0..95 float4-column
    int sel = q >> 5;                 // 0: x[src], 1: x[dst], 2: edge embed
    int off = (q & 31) * 4;
    const float* rp;
    if      (sel == 0) rp = x + (size_t)srcs[el] * 128;
    else if (sel == 1) rp = x + (size_t)dsts[el] * 128;
    else               rp = gbase[el];
    float4 vv = *(const float4*)(rp + off);
    *(float4*)&trip[el * ROWP + q * 4] = vv;
  }
  __syncthreads();

  // WMMA: c[32 edges, this wave's 16 cols] = triple @ w1^T, fp32 accumulate
  const int ncol  = (wave << 4) + (lane & 15);   // output feature column
  const int khalf = (lane >> 4) << 1;            // lanes 16-31 hold K=+2,+3
  const int m     = lane & 15;
  const float* w1row = w1 + (size_t)ncol * KDIM; // B[k][n] = w1[n][k]
  v8f acc0 = {}; v8f acc1 = {};

  #pragma unroll 4
  for (int ks = 0; ks < KDIM / 4; ++ks) {
    int k0 = ks * 4 + khalf;
    v2f b  = *(const v2f*)(w1row + k0);                    // global b64 (L2-hot)
    v2f a0 = *(const v2f*)&trip[m * ROWP + k0];            // ds_read_b64
    v2f a1 = *(const v2f*)&trip[(16 + m) * ROWP + k0];
    acc0 = __builtin_amdgcn_wmma_f32_16x16x4_f32(false, a0, false, b, (short)0, acc0, false, false);
    acc1 = __builtin_amdgcn_wmma_f32_16x16x4_f32(false, a1, false, b, (short)0, acc1, false, false);
  }
  __syncthreads();                      // all waves done reading trip -> reuse as c tile

  float* c_lds = trip;                  // [32][CPAD]
  {
    int m0 = (lane >> 4) << 3;          // C layout: lanes 16-31 hold M+8
    #pragma unroll
    for (int r = 0; r < 8; ++r) {
      c_lds[(m0 + r) * CPAD + ncol]      = acc0[r];
      c_lds[(16 + m0 + r) * CPAD + ncol] = acc1[r];
    }
  }
  __syncthreads();

  // b = leaky_relu(c . w2), bexp = exp(b); 8 threads per edge do partial dots
  {
    int e = tid >> 3, ch = tid & 7;
    float s = 0.f;
    #pragma unroll
    for (int j = 0; j < 16; ++j) s += c_lds[e * CPAD + ch * 16 + j] * w2[ch * 16 + j];
    partial[e * 8 + ch] = s;
  }
  __syncthreads();
  if (tid < TILE_E) {
    float dot = 0.f;
    #pragma unroll
    for (int j = 0; j < 8; ++j) dot += partial[tid * 8 + j];
    float bb = dot > 0.f ? dot : 0.2f * dot;               // leaky relu (slope 0.2)
    float be = evalid[tid] ? expf(bb) : 0.f;
    bexp[tid] = be;
    if (evalid[tid])
      __hip_atomic_fetch_add(&denom[srcs[tid]], be, __ATOMIC_RELAXED, __HIP_MEMORY_SCOPE_AGENT);
  }
  __syncthreads();

  // num[src] += bexp * c   (32 edges x 128 cols = 4096 atomics, 16 per thread)
  #pragma unroll
  for (int i = 0; i < 16; ++i) {
    int idx = tid + (i << 8);
    int e = idx >> 7, col = idx & 127;
    float v = bexp[e] * c_lds[e * CPAD + col];
    __hip_atomic_fetch_add(&out[(size_t)srcs[e] * 128 + col], v,
                           __ATOMIC_RELAXED, __HIP_MEMORY_SCOPE_AGENT);
  }
}

// ---------------- finalize: out = num / (denom + 1e-12) ----------------
__global__ __launch_bounds__(256) void gat_final(float* __restrict__ out,
                                                 const float* __restrict__ denom,
                                                 int n) {
  int i = blockIdx.x * 256 + threadIdx.x;
  if (i < n) out[i] = out[i] / (denom[i >> 7] + 1e-12f);
}

extern "C" void kernel_launch(void* const* d_in, const int* in_sizes, int n_in,
                              void* d_out, int out_size, void* d_ws, size_t ws_size,
                              hipStream_t stream) {
  const float* x       = (const float*)d_in[0];
  const int*   edges   = (const int*)  d_in[1];
  const float* eembed  = (const float*)d_in[2];
  const int*   nedges  = (const int*)  d_in[3];
  const float* neembed = (const float*)d_in[4];
  const float* w1      = (const float*)d_in[5];
  const float* w2      = (const float*)d_in[6];
  float* out   = (float*)d_out;
  float* denom = (float*)d_ws;          // N floats of scratch

  int E1 = in_sizes[1] / 2;
  int E2 = in_sizes[3] / 2;
  int Nn = in_sizes[0] / 128;
  int E  = E1 + E2;

  gat_zero<<<1024, 256, 0, stream>>>(out, denom, Nn * 128, Nn);
  gat_main<<<(E + TILE_E - 1) / TILE_E, 256, 0, stream>>>(
      x, edges, eembed, nedges, neembed, w1, w2, out, denom, E1, E2);
  gat_final<<<(Nn * 128 + 255) / 256, 256, 0, stream>>>(out, denom, Nn * 128);
}